// Transone_36309653520526
// MI455X (gfx1250) — compile-verified
//
#include <hip/hip_runtime.h>

#define B_    16
#define S_    1024
#define NIN   7
#define NH    28
#define HEADS 7

typedef __attribute__((ext_vector_type(2))) float v2f;
typedef __attribute__((ext_vector_type(8))) float v8f;

// ---- 16-lane (half-wave) xor reductions.
// Prefer v_permlane16_b32 (pure VALU) over __shfl_xor (lowers to ds_bpermute).
#if __has_builtin(__builtin_amdgcn_permlane16)
static __device__ __forceinline__ float pl16(float v, unsigned s0, unsigned s1) {
  int i = __float_as_int(v);
  int r = __builtin_amdgcn_permlane16(i, i, s0, s1, false, false);
  return __int_as_float(r);
}
static __device__ __forceinline__ float h16_max(float v) {
  v = fmaxf(v, pl16(v, 0x67452301u, 0xEFCDAB89u));  // xor 1
  v = fmaxf(v, pl16(v, 0x54761032u, 0xDCFE98BAu));  // xor 2
  v = fmaxf(v, pl16(v, 0x32107654u, 0xBA98FEDCu));  // xor 4
  v = fmaxf(v, pl16(v, 0xFEDCBA98u, 0x76543210u));  // xor 8
  return v;
}
static __device__ __forceinline__ float h16_sum(float v) {
  v += pl16(v, 0x67452301u, 0xEFCDAB89u);
  v += pl16(v, 0x54761032u, 0xDCFE98BAu);
  v += pl16(v, 0x32107654u, 0xBA98FEDCu);
  v += pl16(v, 0xFEDCBA98u, 0x76543210u);
  return v;
}
#else
static __device__ __forceinline__ float h16_max(float v) {
  v = fmaxf(v, __shfl_xor(v, 1, 16));
  v = fmaxf(v, __shfl_xor(v, 2, 16));
  v = fmaxf(v, __shfl_xor(v, 4, 16));
  v = fmaxf(v, __shfl_xor(v, 8, 16));
  return v;
}
static __device__ __forceinline__ float h16_sum(float v) {
  v += __shfl_xor(v, 1, 16);
  v += __shfl_xor(v, 2, 16);
  v += __shfl_xor(v, 4, 16);
  v += __shfl_xor(v, 8, 16);
  return v;
}
#endif

// ---------------- Kernel 1: QKV projections (7 -> 28, x3) ----------------
__global__ __launch_bounds__(256) void qkv_proj(
    const float* __restrict__ x,
    const float* __restrict__ Wq, const float* __restrict__ bq,
    const float* __restrict__ Wk, const float* __restrict__ bk,
    const float* __restrict__ Wv, const float* __restrict__ bv,
    float* __restrict__ Q, float* __restrict__ K, float* __restrict__ V) {
  int idx = blockIdx.x * blockDim.x + threadIdx.x;
  if (idx >= B_ * S_) return;
  float xi[NIN];
#pragma unroll
  for (int i = 0; i < NIN; ++i) xi[i] = x[(size_t)idx * NIN + i];
#pragma unroll 4
  for (int j = 0; j < NH; ++j) {
    float q = bq[j], k = bk[j], v = bv[j];
#pragma unroll
    for (int i = 0; i < NIN; ++i) {
      q = fmaf(xi[i], Wq[i * NH + j], q);
      k = fmaf(xi[i], Wk[i * NH + j], k);
      v = fmaf(xi[i], Wv[i * NH + j], v);
    }
    Q[(size_t)idx * NH + j] = q;
    K[(size_t)idx * NH + j] = k;
    V[(size_t)idx * NH + j] = v;
  }
}

// ------- Kernel 2: per-head flash attention; scores via WMMA f32 16x16x4,
// P@V + row-sums accumulated per-lane in VALU (no LDS, no barriers). -------
// grid = (S/16, HEADS, B), block = 32 (one wave)
__global__ __launch_bounds__(32) void attn1(
    const float* __restrict__ Q, const float* __restrict__ K,
    const float* __restrict__ V, float* __restrict__ A1) {
  const int qt = blockIdx.x, h = blockIdx.y, b = blockIdx.z;
  const int lane = threadIdx.x;
  const int half = lane >> 4;
  const int lid  = lane & 15;
  const int q0 = qt * 16;
  const size_t base = (size_t)b * S_ * NH;

  // Q tile in A-layout: lane holds (m=lid, kdim=2*half+{0,1}) -> contiguous b64
  v2f qa;
  {
    const float* qp = Q + base + (size_t)(q0 + lid) * NH + 4 * h + 2 * half;
    qa.x = qp[0]; qa.y = qp[1];
  }

  float mrow[8];          // running row max (row r+8*half)
  float acc[8][4];        // per-lane partial O (this lane's k columns only)
  float accl[8];          // per-lane partial row-sum of p
#pragma unroll
  for (int r = 0; r < 8; ++r) {
    mrow[r] = -3.0e38f; accl[r] = 0.0f;
    acc[r][0] = acc[r][1] = acc[r][2] = acc[r][3] = 0.0f;
  }

  for (int k0 = 0; k0 < S_; k0 += 64) {  // 4 chunks of 16 per iteration
    v8f s[4];
    float vv[4][4];
#pragma unroll
    for (int c = 0; c < 4; ++c) {
      const int kc = k0 + 16 * c;
      v2f kb;  // Kt B-layout: (kdim=2*half+{0,1}, n=lid)
      {
        const float* kp = K + base + (size_t)(kc + lid) * NH + 4 * h + 2 * half;
        kb.x = kp[0]; kb.y = kp[1];
      }
      v8f cz = {};
      s[c] = __builtin_amdgcn_wmma_f32_16x16x4_f32(
          false, qa, false, kb, (short)0, cz, false, false);
      float4 vr = *(const float4*)(V + base + (size_t)(kc + lid) * NH + 4 * h);
      vv[c][0] = vr.x; vv[c][1] = vr.y; vv[c][2] = vr.z; vv[c][3] = vr.w;
    }
    if (k0 + 64 < S_) {  // stream next iteration's K/V through the cache path
      __builtin_prefetch(K + base + (size_t)(k0 + 64 + lid) * NH + 4 * h, 0, 0);
      __builtin_prefetch(V + base + (size_t)(k0 + 64 + lid) * NH + 4 * h, 0, 0);
    }
#pragma unroll
    for (int r = 0; r < 8; ++r) {
      // one lane-reduction per row per 4 chunks
      float sm = fmaxf(fmaxf(s[0][r], s[1][r]), fmaxf(s[2][r], s[3][r])) * 0.5f;
      float mnew = fmaxf(mrow[r], h16_max(sm));
      float alpha = __expf(mrow[r] - mnew);
      mrow[r] = mnew;
      float p0 = __expf(fmaf(s[0][r], 0.5f, -mnew));
      float p1 = __expf(fmaf(s[1][r], 0.5f, -mnew));
      float p2 = __expf(fmaf(s[2][r], 0.5f, -mnew));
      float p3 = __expf(fmaf(s[3][r], 0.5f, -mnew));
      accl[r] = fmaf(accl[r], alpha, (p0 + p1) + (p2 + p3));
#pragma unroll
      for (int d = 0; d < 4; ++d) {
        float a = acc[r][d] * alpha;
        a = fmaf(p0, vv[0][d], a);
        a = fmaf(p1, vv[1][d], a);
        a = fmaf(p2, vv[2][d], a);
        a = fmaf(p3, vv[3][d], a);
        acc[r][d] = a;
      }
    }
  }

  // final cross-lane reduction (once), then write 16x4 head output
#pragma unroll
  for (int r = 0; r < 8; ++r) {
    accl[r] = h16_sum(accl[r]);
#pragma unroll
    for (int d = 0; d < 4; ++d) acc[r][d] = h16_sum(acc[r][d]);
  }
  if (lid < 4) {
#pragma unroll
    for (int r = 0; r < 8; ++r) {
      float o = acc[r][0];
      if (lid == 1) o = acc[r][1];
      if (lid == 2) o = acc[r][2];
      if (lid == 3) o = acc[r][3];
      A1[base + (size_t)(q0 + r + 8 * half) * NH + 4 * h + lid] = o / accl[r];
    }
  }
}

// ------ Kernel 3: LayerNorm + FFN + ReLU + residual, Q2/K2 projections ------
__global__ __launch_bounds__(256) void mid_stage(
    const float* __restrict__ A1, const float* __restrict__ x,
    const float* __restrict__ ln_w, const float* __restrict__ ln_b,
    const float* __restrict__ W1, const float* __restrict__ b1,
    const float* __restrict__ Wq2, const float* __restrict__ bq2,
    const float* __restrict__ Wk2, const float* __restrict__ bk2,
    float* __restrict__ out, float* __restrict__ Q2, float* __restrict__ K2) {
  int idx = blockIdx.x * blockDim.x + threadIdx.x;
  if (idx >= B_ * S_) return;
  float a[NH];
  float mu = 0.0f;
#pragma unroll 4
  for (int j = 0; j < NH; ++j) { a[j] = A1[(size_t)idx * NH + j]; mu += a[j]; }
  mu *= (1.0f / NH);
  float var = 0.0f;
#pragma unroll 4
  for (int j = 0; j < NH; ++j) { float d = a[j] - mu; var = fmaf(d, d, var); }
  var *= (1.0f / NH);
  float inv = __frsqrt_rn(var + 1e-5f);
#pragma unroll 4
  for (int j = 0; j < NH; ++j) a[j] = fmaf((a[j] - mu) * inv, ln_w[j], ln_b[j]);

  float hf[NIN];
#pragma unroll
  for (int i = 0; i < NIN; ++i) {
    float t = b1[i];
#pragma unroll 4
    for (int j = 0; j < NH; ++j) t = fmaf(a[j], W1[j * NIN + i], t);
    hf[i] = fmaxf(t, 0.0f) + x[(size_t)idx * NIN + i];
    out[(size_t)idx * NIN + i] = hf[i];
  }
#pragma unroll 4
  for (int j = 0; j < NH; ++j) {
    float q = bq2[j], k = bk2[j];
#pragma unroll
    for (int i = 0; i < NIN; ++i) {
      q = fmaf(hf[i], Wq2[i * NH + j], q);
      k = fmaf(hf[i], Wk2[i * NH + j], k);
    }
    Q2[(size_t)idx * NH + j] = q;
    K2[(size_t)idx * NH + j] = k;
  }
}

// ---- Kernel 4: attn2 = softmax(Q2 K2^T / sqrt(7)). Score loop is pure
// WMMA + ds_store; softmax fully deferred over the 64KB LDS stage. ----
// grid = (S/16, B), block = 32 (one wave)
#define SSTR 1024
__global__ __launch_bounds__(32) void attn2_kernel(
    const float* __restrict__ Q2, const float* __restrict__ K2,
    float* __restrict__ attn2) {
  const int qt = blockIdx.x, b = blockIdx.y;
  const int lane = threadIdx.x;
  const int half = lane >> 4;
  const int lid  = lane & 15;
  const int q0 = qt * 16;
  const size_t base = (size_t)b * S_ * NH;

  __shared__ float sc[16 * SSTR];  // 64 KB raw scores

  // Q2 tile: 7 A-fragments covering K=28
  v2f qa[7];
#pragma unroll
  for (int c = 0; c < 7; ++c) {
    const float* qp = Q2 + base + (size_t)(q0 + lid) * NH + 4 * c + 2 * half;
    qa[c].x = qp[0]; qa[c].y = qp[1];
  }

  const float scale = 0.3779644730092272f;  // 1/sqrt(7)

  for (int k0 = 0; k0 < S_; k0 += 16) {
    v8f s = {};
#pragma unroll
    for (int c = 0; c < 7; ++c) {
      v2f kb;
      const float* kp = K2 + base + (size_t)(k0 + lid) * NH + 4 * c + 2 * half;
      kb.x = kp[0]; kb.y = kp[1];
      s = __builtin_amdgcn_wmma_f32_16x16x4_f32(
          false, qa[c], false, kb, (short)0, s, false, false);
    }
    if (k0 + 16 < S_)
      __builtin_prefetch(K2 + base + (size_t)(k0 + 16 + lid) * NH, 0, 0);
#pragma unroll
    for (int r = 0; r < 8; ++r)
      sc[(r + 8 * half) * SSTR + k0 + lid] = s[r] * scale;
  }
  __syncthreads();

  // deferred softmax over LDS: max pass, sum pass, normalize+write pass.
  // lane covers cols {64j + 4*lid .. +3} of rows r+8*half (float4, conflict-free).
  float mrow[8], lrow[8];
#pragma unroll
  for (int r = 0; r < 8; ++r) {
    const float4* sp = (const float4*)&sc[(r + 8 * half) * SSTR];
    float mx = -3.0e38f;
#pragma unroll 4
    for (int j = 0; j < 16; ++j) {
      float4 t = sp[16 * j + lid];
      mx = fmaxf(mx, fmaxf(fmaxf(t.x, t.y), fmaxf(t.z, t.w)));
    }
    mrow[r] = h16_max(mx);
  }
#pragma unroll
  for (int r = 0; r < 8; ++r) {
    const float4* sp = (const float4*)&sc[(r + 8 * half) * SSTR];
    float mm = mrow[r];
    float ssum = 0.0f;
#pragma unroll 4
    for (int j = 0; j < 16; ++j) {
      float4 t = sp[16 * j + lid];
      ssum += __expf(t.x - mm) + __expf(t.y - mm) +
              __expf(t.z - mm) + __expf(t.w - mm);
    }
    lrow[r] = h16_sum(ssum);
  }
  float* op = attn2 + ((size_t)b * S_ + q0) * S_;
#pragma unroll
  for (int r = 0; r < 8; ++r) {
    const int m = r + 8 * half;
    const float4* sp = (const float4*)&sc[m * SSTR];
    float4* dp = (float4*)&op[(size_t)m * S_];
    float mm = mrow[r];
    float li = 1.0f / lrow[r];
#pragma unroll 4
    for (int j = 0; j < 16; ++j) {
      float4 t = sp[16 * j + lid];
      float4 o;
      o.x = __expf(t.x - mm) * li;
      o.y = __expf(t.y - mm) * li;
      o.z = __expf(t.z - mm) * li;
      o.w = __expf(t.w - mm) * li;
      dp[16 * j + lid] = o;  // 256B contiguous per half-wave per instruction
    }
  }
}

extern "C" void kernel_launch(void* const* d_in, const int* in_sizes, int n_in,
                              void* d_out, int out_size, void* d_ws, size_t ws_size,
                              hipStream_t stream) {
  const float* x    = (const float*)d_in[0];
  const float* Wq   = (const float*)d_in[1];
  const float* bq   = (const float*)d_in[2];
  const float* Wk   = (const float*)d_in[3];
  const float* bk   = (const float*)d_in[4];
  const float* Wv   = (const float*)d_in[5];
  const float* bv   = (const float*)d_in[6];
  const float* ln_w = (const float*)d_in[7];
  const float* ln_b = (const float*)d_in[8];
  const float* W1   = (const float*)d_in[9];
  const float* b1   = (const float*)d_in[10];
  const float* Wq2  = (const float*)d_in[11];
  const float* bq2  = (const float*)d_in[12];
  const float* Wk2  = (const float*)d_in[13];
  const float* bk2  = (const float*)d_in[14];

  const size_t BSH = (size_t)B_ * S_ * NH;
  float* ws = (float*)d_ws;
  float* Q  = ws + 0 * BSH;
  float* K  = ws + 1 * BSH;
  float* V  = ws + 2 * BSH;
  float* A1 = ws + 3 * BSH;
  float* Q2 = ws + 4 * BSH;
  float* K2 = ws + 5 * BSH;

  float* out   = (float*)d_out;                 // [B,S,7]
  float* attn2 = out + (size_t)B_ * S_ * NIN;   // [B,S,S]

  const int rows = B_ * S_;
  qkv_proj<<<(rows + 255) / 256, 256, 0, stream>>>(x, Wq, bq, Wk, bk, Wv, bv, Q, K, V);
  attn1<<<dim3(S_ / 16, HEADS, B_), 32, 0, stream>>>(Q, K, V, A1);
  mid_stage<<<(rows + 255) / 256, 256, 0, stream>>>(A1, x, ln_w, ln_b, W1, b1,
                                                    Wq2, bq2, Wk2, bk2, out, Q2, K2);
  attn2_kernel<<<dim3(S_ / 16, B_), 32, 0, stream>>>(Q2, K2, attn2);
}